// TwoLayerAttentionClassifier_39170101740308
// MI455X (gfx1250) — compile-verified
//
#include <hip/hip_runtime.h>

#define EMBED 768
#define HID   1024
#define NCLS  20
#define BB    64
#define KK    32
#define TT    64
#define NEG_INF -1e30f

typedef __attribute__((ext_vector_type(2))) float        v2f;
typedef __attribute__((ext_vector_type(8))) float        v8f;
typedef __attribute__((ext_vector_type(4))) unsigned int v4u;
typedef __attribute__((ext_vector_type(8))) int          v8i;
typedef __attribute__((ext_vector_type(4))) int          v4i;

// ---------------------------------------------------------------------------
// TDM descriptor: DMA `rows` x EMBED fp32 tile (row-major, contiguous) from
// global memory into LDS at byte offset `lds_off`.  D# layout per CDNA5 ISA
// ch.8: group0 = count/lds_addr/global_addr/type, group1 = sizes/strides.
// Groups 2/3 zero (tensor is 2D).  This toolchain uses the 6-arg builtin
// (extra int32x8 group, zero-filled) + cpol.
// ---------------------------------------------------------------------------
__device__ __forceinline__ void tdm_load_rows_to_lds(const float* gsrc,
                                                     unsigned lds_off,
                                                     unsigned rows)
{
    const unsigned long long ga = (unsigned long long)(size_t)gsrc;
    v4u g0;
    g0.x = 1u;                                        // count=1 (valid), user mode
    g0.y = lds_off;                                   // lds_addr (bytes)
    g0.z = (unsigned)(ga & 0xFFFFFFFFull);            // global_addr[31:0]
    g0.w = (unsigned)(ga >> 32) | (2u << 30);         // global_addr[56:32] | type=2

    v8i g1;
    g1[0] = (int)(2u << 16);                          // wg_mask=0, data_size=2 (4B)
    g1[1] = (int)((EMBED & 0xFFFFu) << 16);           // tensor_dim0[15:0] @ bits63:48
    g1[2] = (int)(((EMBED >> 16) & 0xFFFFu) | (rows << 16)); // dim0 hi | tensor_dim1 lo
    g1[3] = (int)(((rows >> 16) & 0xFFFFu) | ((unsigned)EMBED << 16)); // dim1 hi | tile_dim0
    g1[4] = (int)(rows & 0xFFFFu);                    // tile_dim1 | tile_dim2=0
    g1[5] = (int)EMBED;                               // tensor_dim0_stride[31:0]
    g1[6] = 0;                                        // stride hi | dim1_stride lo
    g1[7] = 0;
    v4i g2 = {}, g3 = {};
    v8i g4 = {};
    __builtin_amdgcn_tensor_load_to_lds(g0, g1, g2, g3, g4, 0);
}

// ---------------------------------------------------------------------------
// Kernel 1: token-level masked attention pooling, one block per (b,k) pair.
// The 64x768 fp32 row (192 KB) is DMA'd into LDS by the Tensor Data Mover in
// two 32-token descriptors so score compute overlaps the second transfer.
// Both passes (scores + weighted sum) then run entirely out of LDS: one HBM
// read of the 403 MB tensor total.
// ---------------------------------------------------------------------------
__global__ void __launch_bounds__(256)
token_pool_kernel(const float* __restrict__ x,
                  const unsigned char* __restrict__ kp_mask,
                  const unsigned char* __restrict__ token_mask,
                  const float* __restrict__ w_token,
                  const float* __restrict__ w_kp,
                  float* __restrict__ kp_vec,
                  float* __restrict__ kp_scores)
{
    __shared__ float s_x[TT * EMBED];   // 192 KB row tile (TDM destination)
    __shared__ float s_score[TT];
    __shared__ float s_w[TT];
    __shared__ float s_red[8];

    const int bk   = blockIdx.x;        // 0..2047  (b*32 + k)
    const int tid  = threadIdx.x;       // 0..255
    const int lane = tid & 31;
    const int wave = tid >> 5;          // 0..7

    const float* xb = x + (size_t)bk * TT * EMBED;
    const unsigned char* tmask = token_mask + (size_t)bk * TT;

    // ---- issue TDM DMA: two half-row descriptors (in-order completion) ----
    if (wave == 0) {
        const unsigned lds_base = (unsigned)(size_t)(s_x); // LDS aperture: off in [31:0]
        tdm_load_rows_to_lds(xb, lds_base, 32u);
        tdm_load_rows_to_lds(xb + 32 * EMBED, lds_base + 32u * EMBED * 4u, 32u);
        __builtin_amdgcn_s_wait_tensorcnt(1);   // first half landed
    }
    __syncthreads();

    // ---- pass A: token scores from LDS (wave w -> 4 tokens per half) ----
    const float4* wv = (const float4*)w_token;
#pragma unroll
    for (int half = 0; half < 2; ++half) {
        if (half == 1) {
            if (wave == 0) __builtin_amdgcn_s_wait_tensorcnt(0);
            __syncthreads();                    // second half landed, visible to all
        }
        for (int t = half * 32 + wave * 4; t < half * 32 + wave * 4 + 4; ++t) {
            const float4* row = (const float4*)(s_x + t * EMBED);
            float acc = 0.f;
#pragma unroll
            for (int j = 0; j < 6; ++j) {
                const int e4 = lane + j * 32;   // 192 float4 per token
                float4 v = row[e4];
                float4 w = wv[e4];
                acc += v.x * w.x + v.y * w.y + v.z * w.z + v.w * w.w;
            }
            for (int off = 16; off > 0; off >>= 1)
                acc += __shfl_xor(acc, off, 32);
            if (lane == 0)
                s_score[t] = tmask[t] ? acc : NEG_INF;
        }
    }
    __syncthreads();

    // ---- softmax over 64 token scores (redundant per-thread, cheap) ----
    float mx = NEG_INF;
    for (int t = 0; t < TT; ++t) mx = fmaxf(mx, s_score[t]);
    float sum = 0.f;
    for (int t = 0; t < TT; ++t) sum += __expf(s_score[t] - mx);
    if (tid < TT) s_w[tid] = __expf(s_score[tid] - mx) / sum;
    __syncthreads();

    // ---- pass B: weighted sum over tokens, entirely from LDS ----
    const int e0 = tid, e1 = tid + 256, e2 = tid + 512;
    float a0 = 0.f, a1 = 0.f, a2 = 0.f;
    for (int t = 0; t < TT; ++t) {
        const float  wt  = s_w[t];
        const float* row = s_x + t * EMBED;
        a0 += wt * row[e0];
        a1 += wt * row[e1];
        a2 += wt * row[e2];
    }
    const bool km = kp_mask[bk] != 0;
    if (!km) { a0 = a1 = a2 = 0.f; }

    float* out = kp_vec + (size_t)bk * EMBED;
    out[e0] = a0; out[e1] = a1; out[e2] = a2;

    // ---- fused kp-level score: kp_vec . w_kp ----
    float p = a0 * w_kp[e0] + a1 * w_kp[e1] + a2 * w_kp[e2];
    for (int off = 16; off > 0; off >>= 1)
        p += __shfl_xor(p, off, 32);
    if (lane == 0) s_red[wave] = p;
    __syncthreads();
    if (tid == 0) {
        float s = 0.f;
        for (int i = 0; i < 8; ++i) s += s_red[i];
        kp_scores[bk] = km ? s : NEG_INF;
    }
}

// ---------------------------------------------------------------------------
// Kernel 2: kp-level softmax + pooling, one block per batch element.
// ---------------------------------------------------------------------------
__global__ void kp_pool_kernel(const float* __restrict__ kp_vec,
                               const float* __restrict__ kp_scores,
                               float* __restrict__ pooled)
{
    __shared__ float s_s[KK];
    __shared__ float s_w[KK];
    const int b = blockIdx.x;
    const int tid = threadIdx.x;

    if (tid < KK) s_s[tid] = kp_scores[b * KK + tid];
    __syncthreads();

    float mx = NEG_INF;
    for (int k = 0; k < KK; ++k) mx = fmaxf(mx, s_s[k]);
    float sum = 0.f;
    for (int k = 0; k < KK; ++k) sum += __expf(s_s[k] - mx);
    if (tid < KK) s_w[tid] = __expf(s_s[tid] - mx) / sum;
    __syncthreads();

    const int e0 = tid, e1 = tid + 256, e2 = tid + 512;
    const float* base = kp_vec + (size_t)b * KK * EMBED;
    float a0 = 0.f, a1 = 0.f, a2 = 0.f;
    for (int k = 0; k < KK; ++k) {
        const float  wt  = s_w[k];
        const float* row = base + (size_t)k * EMBED;
        a0 += wt * row[e0];
        a1 += wt * row[e1];
        a2 += wt * row[e2];
    }
    float* out = pooled + (size_t)b * EMBED;
    out[e0] = a0; out[e1] = a1; out[e2] = a2;
}

// ---------------------------------------------------------------------------
// Kernel 3a: h = relu(pooled @ W1 + b1) via V_WMMA_F32_16X16X4_F32.
// One wave (32 lanes) computes one 16x16 output tile; K loop steps by 4.
// A layout (16x4 f32): lanes 0-15 -> K={0,1}; lanes 16-31 -> K={2,3}.
// B layout (4x16 f32): VGPR0 = rows K={0|2}, VGPR1 = rows K={1|3}, N = lane&15.
// C/D layout (16x16 f32): VGPR i -> M = i + 8*(lane>>4), N = lane&15.
// ---------------------------------------------------------------------------
__global__ void fc1_wmma_kernel(const float* __restrict__ pooled, // [64,768]
                                const float* __restrict__ W1,     // [768,1024]
                                const float* __restrict__ b1,     // [1024]
                                float* __restrict__ h)            // [64,1024]
{
    const int lane = threadIdx.x;        // 0..31 (one wave per block)
    const int n0   = blockIdx.x * 16;    // output column tile
    const int m0   = blockIdx.y * 16;    // output row tile
    const int half = lane >> 4;          // 0 or 1
    const int l15  = lane & 15;

    v8f c = {};
    const float* Arow = pooled + (size_t)(m0 + l15) * EMBED;

    for (int k = 0; k < EMBED; k += 4) {
        const int kk = k + half * 2;
        v2f a, bm;
        a.x  = Arow[kk];
        a.y  = Arow[kk + 1];
        bm.x = W1[(size_t)kk * HID + n0 + l15];
        bm.y = W1[(size_t)(kk + 1) * HID + n0 + l15];
        c = __builtin_amdgcn_wmma_f32_16x16x4_f32(
                /*neg_a=*/false, a, /*neg_b=*/false, bm,
                /*c_mod=*/(short)0, c, /*reuse_a=*/false, /*reuse_b=*/false);
    }

#pragma unroll
    for (int i = 0; i < 8; ++i) {
        const int m = m0 + i + half * 8;
        const int n = n0 + l15;
        const float v = c[i] + b1[n];
        h[(size_t)m * HID + n] = fmaxf(v, 0.f);
    }
}

// ---------------------------------------------------------------------------
// Kernel 3b: logits = h @ W2 + b2. One block per batch row; 20 warps, one per
// class; warp-strided dot over HID with shfl reduction.
// ---------------------------------------------------------------------------
__global__ void fc2_kernel(const float* __restrict__ h,    // [64,1024]
                           const float* __restrict__ W2,   // [1024,20]
                           const float* __restrict__ b2,   // [20]
                           float* __restrict__ logits)     // [64,20]
{
    const int b    = blockIdx.x;
    const int lane = threadIdx.x;  // 0..31
    const int cls  = threadIdx.y;  // 0..19

    const float* hr = h + (size_t)b * HID;
    float acc = 0.f;
    for (int k = lane; k < HID; k += 32)
        acc += hr[k] * W2[(size_t)k * NCLS + cls];
    for (int off = 16; off > 0; off >>= 1)
        acc += __shfl_xor(acc, off, 32);
    if (lane == 0)
        logits[b * NCLS + cls] = acc + b2[cls];
}

// ---------------------------------------------------------------------------
extern "C" void kernel_launch(void* const* d_in, const int* in_sizes, int n_in,
                              void* d_out, int out_size, void* d_ws, size_t ws_size,
                              hipStream_t stream)
{
    const float*         x          = (const float*)d_in[0];
    const unsigned char* kp_mask    = (const unsigned char*)d_in[1];
    const unsigned char* token_mask = (const unsigned char*)d_in[2];
    const float*         w_token    = (const float*)d_in[3];
    const float*         w_kp       = (const float*)d_in[4];
    const float*         W1         = (const float*)d_in[5];
    const float*         b1         = (const float*)d_in[6];
    const float*         W2         = (const float*)d_in[7];
    const float*         b2         = (const float*)d_in[8];
    float*               logits     = (float*)d_out;

    // workspace layout (fp32):
    //   kp_vec    : 64*32*768   = 1,572,864
    //   kp_scores : 64*32       = 2,048
    //   pooled    : 64*768      = 49,152
    //   h         : 64*1024     = 65,536      (total ~6.75 MB)
    float* kp_vec    = (float*)d_ws;
    float* kp_scores = kp_vec + (size_t)BB * KK * EMBED;
    float* pooled    = kp_scores + BB * KK;
    float* h         = pooled + (size_t)BB * EMBED;

    token_pool_kernel<<<BB * KK, 256, 0, stream>>>(x, kp_mask, token_mask,
                                                   w_token, w_kp, kp_vec, kp_scores);
    kp_pool_kernel<<<BB, 256, 0, stream>>>(kp_vec, kp_scores, pooled);
    fc1_wmma_kernel<<<dim3(HID / 16, BB / 16), 32, 0, stream>>>(pooled, W1, b1, h);
    fc2_kernel<<<BB, dim3(32, NCLS), 0, stream>>>(h, W2, b2, logits);
}